// HoGANet_89661737271572
// MI455X (gfx1250) — compile-verified
//
#include <hip/hip_runtime.h>
#include <hip/hip_bf16.h>
#include <math.h>

#define NN 50000
#define INF_ 256
#define FF 64
#define HH 4
#define MM 3
#define EE 800000
#define OUTF 16
#define HF (HH * FF)   // 256

typedef __attribute__((ext_vector_type(2))) float v2f;
typedef __attribute__((ext_vector_type(8))) float v8f;

// ---------------- helpers ----------------

__device__ __forceinline__ float leaky02(float x) {
    return x > 0.0f ? x : 0.2f * x;
}

__device__ __forceinline__ float elu1(float x) {
    return x > 0.0f ? x : (expf(x) - 1.0f);
}

// float atomic max via sign-split integer atomics (portable, correct for all signs)
__device__ __forceinline__ void atomicMaxFloat(float* addr, float value) {
    if (value >= 0.0f) {
        atomicMax((int*)addr, __float_as_int(value));
    } else {
        atomicMin((unsigned int*)addr, __float_as_uint(value));
    }
}

// ---------------- K1: feat[m] = h @ W_fc[m]  (fp32 WMMA 16x16x4) ----------------
// grid: (N/16, 2, M), block: 256 (8 waves). wave w of blockIdx.y half covers col tile.
__global__ void gemm_feat_kernel(const float* __restrict__ hsrc,
                                 const float* __restrict__ Wfc,
                                 float* __restrict__ feat) {
    const int lane = threadIdx.x & 31;
    const int wv   = threadIdx.x >> 5;
    const int rowTile = blockIdx.x;                 // 0..3124
    const int colTile = blockIdx.y * 8 + wv;        // 0..15
    const int m = blockIdx.z;

    const int half = lane >> 4;     // K sub-pair select
    const int lrow = lane & 15;

    const float* A = hsrc;                                        // [N, IN]
    const float* B = Wfc + (long long)m * INF_ * HF;              // [IN, HF]
    const int arow = rowTile * 16 + lrow;
    const int bcol = colTile * 16 + lrow;

    v8f c = {};
    for (int k0 = 0; k0 < INF_; k0 += 4) {
        const int ka = k0 + 2 * half;
        // A fragment: lane holds A[arow][ka], A[arow][ka+1]
        const float2 av = *(const float2*)(A + (long long)arow * INF_ + ka);
        v2f a; a[0] = av.x; a[1] = av.y;
        // B fragment: lane holds B[ka][bcol], B[ka+1][bcol]
        v2f b;
        b[0] = B[(long long)ka * HF + bcol];
        b[1] = B[(long long)(ka + 1) * HF + bcol];
        c = __builtin_amdgcn_wmma_f32_16x16x4_f32(false, a, false, b,
                                                  (short)0, c, false, false);
    }

    float* Fm = feat + (long long)m * NN * HF;
#pragma unroll
    for (int r = 0; r < 8; ++r) {
        const int n = rowTile * 16 + r + 8 * half;
        Fm[(long long)n * HF + bcol] = c[r];
    }
}

// ---------------- K2: el/er = einsum('nhf,hf->nh') ----------------
// one wave per (m,n); lane covers 8 consecutive feats; 8-lane shuffle reduce per head.
__global__ void el_er_kernel(const float* __restrict__ feat,
                             const float* __restrict__ al,
                             const float* __restrict__ ar,
                             float* __restrict__ el,
                             float* __restrict__ er) {
    const int lane = threadIdx.x & 31;
    const int wv   = threadIdx.x >> 5;
    const long long node = (long long)blockIdx.x * 8 + wv;   // m*N + n
    if (node >= (long long)MM * NN) return;
    const int m = (int)(node / NN);

    const float* frow = feat + node * HF;
    const float* alm  = al + m * HF;   // [H,F] flattened == HF
    const float* arm  = ar + m * HF;

    const int c0 = lane * 8;           // 8 feats per lane; head = lane/8
    float pl = 0.0f, pr = 0.0f;
#pragma unroll
    for (int j = 0; j < 8; j += 4) {
        const float4 fv = *(const float4*)(frow + c0 + j);
        const float4 lv = *(const float4*)(alm + c0 + j);
        const float4 rv = *(const float4*)(arm + c0 + j);
        pl += fv.x * lv.x + fv.y * lv.y + fv.z * lv.z + fv.w * lv.w;
        pr += fv.x * rv.x + fv.y * rv.y + fv.z * rv.z + fv.w * rv.w;
    }
#pragma unroll
    for (int off = 4; off > 0; off >>= 1) {
        pl += __shfl_down(pl, off, 8);
        pr += __shfl_down(pr, off, 8);
    }
    if ((lane & 7) == 0) {
        const int h = lane >> 3;
        el[node * HH + h] = pl;
        er[node * HH + h] = pr;
    }
}

// ---------------- K3: fill ----------------
__global__ void fill_kernel(float* __restrict__ p, float v, long long n) {
    const long long i = (long long)blockIdx.x * blockDim.x + threadIdx.x;
    if (i < n) p[i] = v;
}

// ---------------- K4: segment max over dst ----------------
__global__ void edge_max_kernel(const int* __restrict__ ei,
                                const float* __restrict__ el,
                                const float* __restrict__ er,
                                float* __restrict__ maxb) {
    const long long t = (long long)blockIdx.x * 256 + threadIdx.x;
    if (t >= (long long)MM * EE * HH) return;
    const int h = (int)(t % HH);
    const long long te = t / HH;
    const int e = (int)(te % EE);
    const int m = (int)(te / EE);
    const int src = ei[(long long)m * 2 * EE + e];
    const int dst = ei[(long long)m * 2 * EE + EE + e];
    const float x = el[((long long)m * NN + src) * HH + h] +
                    er[((long long)m * NN + dst) * HH + h];
    atomicMaxFloat(&maxb[((long long)m * NN + dst) * HH + h], leaky02(x));
}

// ---------------- K5: w = exp(e - max[dst]); denom += w ----------------
__global__ void edge_exp_kernel(const int* __restrict__ ei,
                                const float* __restrict__ el,
                                const float* __restrict__ er,
                                const float* __restrict__ maxb,
                                float* __restrict__ wbuf,
                                float* __restrict__ denom) {
    const long long t = (long long)blockIdx.x * 256 + threadIdx.x;
    if (t >= (long long)MM * EE * HH) return;
    const int h = (int)(t % HH);
    const long long te = t / HH;
    const int e = (int)(te % EE);
    const int m = (int)(te / EE);
    const int src = ei[(long long)m * 2 * EE + e];
    const int dst = ei[(long long)m * 2 * EE + EE + e];
    const long long di = ((long long)m * NN + dst) * HH + h;
    const float x = leaky02(el[((long long)m * NN + src) * HH + h] +
                            er[di - ((long long)0)]);  // er at dst
    const float w = expf(x - maxb[di]);
    wbuf[((long long)m * EE + e) * HH + h] = w;
    atomicAdd(&denom[di], w);
}

// ---------------- K6: rst[dst] += (w/denom[dst]) * feat[src] ----------------
__global__ void scatter_kernel(const int* __restrict__ ei,
                               const float* __restrict__ wbuf,
                               const float* __restrict__ denom,
                               const float* __restrict__ feat,
                               float* __restrict__ rst) {
    const long long t = (long long)blockIdx.x * 256 + threadIdx.x;
    if (t >= (long long)MM * EE * HF) return;
    const int c = (int)(t % HF);
    const long long te = t / HF;
    const int e = (int)(te % EE);
    const int m = (int)(te / EE);
    const int h = c >> 6;
    const int src = ei[(long long)m * 2 * EE + e];
    const int dst = ei[(long long)m * 2 * EE + EE + e];
    const float w  = wbuf[((long long)m * EE + e) * HH + h];
    const float dn = denom[((long long)m * NN + dst) * HH + h];
    const float fv = feat[((long long)m * NN + src) * HF + c];
    atomicAdd(&rst[((long long)m * NN + dst) * HF + c], (w / dn) * fv);
}

// ---------------- K7: out[n,m,:] = elu(rst[m,n,:]) @ W_pred + b  (fp32 WMMA) ----------------
// one wave per (rowTile, m); OUT = 16 exactly one WMMA N-tile.
__global__ void pred_gemm_kernel(const float* __restrict__ rst,
                                 const float* __restrict__ Wp,
                                 const float* __restrict__ bp,
                                 float* __restrict__ out) {
    const int lane = threadIdx.x & 31;
    const int wv   = threadIdx.x >> 5;
    const int rowTile = blockIdx.x * 8 + wv;
    if (rowTile >= NN / 16) return;   // wave-uniform exit
    const int m = blockIdx.y;

    const int half = lane >> 4;
    const int lrow = lane & 15;
    const float* A = rst + (long long)m * NN * HF;
    const int arow = rowTile * 16 + lrow;

    v8f c = {};
    for (int k0 = 0; k0 < HF; k0 += 4) {
        const int ka = k0 + 2 * half;
        const float2 av = *(const float2*)(A + (long long)arow * HF + ka);
        v2f a; a[0] = elu1(av.x); a[1] = elu1(av.y);
        v2f b;
        b[0] = Wp[ka * OUTF + lrow];
        b[1] = Wp[(ka + 1) * OUTF + lrow];
        c = __builtin_amdgcn_wmma_f32_16x16x4_f32(false, a, false, b,
                                                  (short)0, c, false, false);
    }
    const float bias = bp[lrow];
#pragma unroll
    for (int r = 0; r < 8; ++r) {
        const int n = rowTile * 16 + r + 8 * half;
        out[(long long)n * (MM * OUTF) + m * OUTF + lrow] = c[r] + bias;
    }
}

// ---------------- launch ----------------
extern "C" void kernel_launch(void* const* d_in, const int* in_sizes, int n_in,
                              void* d_out, int out_size, void* d_ws, size_t ws_size,
                              hipStream_t stream) {
    const float* h_ptr   = (const float*)d_in[0];
    const int*   ei      = (const int*)d_in[1];
    const float* Wfc     = (const float*)d_in[2];
    const float* attn_l  = (const float*)d_in[3];
    const float* attn_r  = (const float*)d_in[4];
    const float* Wpred   = (const float*)d_in[5];
    const float* bpred   = (const float*)d_in[6];
    float* out = (float*)d_out;

    // workspace carve-up (floats)
    float* ws = (float*)d_ws;
    const long long featN  = (long long)MM * NN * HF;   // 38.4M
    const long long nhN    = (long long)MM * NN * HH;   // 0.6M
    const long long wN     = (long long)MM * EE * HH;   // 9.6M
    float* feat  = ws;
    float* rst   = feat  + featN;
    float* el    = rst   + featN;
    float* er    = el    + nhN;
    float* maxb  = er    + nhN;
    float* denom = maxb  + nhN;
    float* wbuf  = denom + nhN;

    // K1: feat GEMM (WMMA f32)
    gemm_feat_kernel<<<dim3(NN / 16, 2, MM), 256, 0, stream>>>(h_ptr, Wfc, feat);

    // K2: attention logits
    {
        const long long nodes = (long long)MM * NN;
        const int blocks = (int)((nodes + 7) / 8);
        el_er_kernel<<<blocks, 256, 0, stream>>>(feat, attn_l, attn_r, el, er);
    }

    // K3: init accumulators
    fill_kernel<<<(int)((featN + 255) / 256), 256, 0, stream>>>(rst, 0.0f, featN);
    fill_kernel<<<(int)((nhN + 255) / 256), 256, 0, stream>>>(maxb, -3.0e38f, nhN);
    fill_kernel<<<(int)((nhN + 255) / 256), 256, 0, stream>>>(denom, 0.0f, nhN);

    // K4/K5: segment softmax over dst
    {
        const long long tot = (long long)MM * EE * HH;
        const int blocks = (int)((tot + 255) / 256);
        edge_max_kernel<<<blocks, 256, 0, stream>>>(ei, el, er, maxb);
        edge_exp_kernel<<<blocks, 256, 0, stream>>>(ei, el, er, maxb, wbuf, denom);
    }

    // K6: weighted scatter of source features
    {
        const long long tot = (long long)MM * EE * HF;  // 614.4M
        const int blocks = (int)((tot + 255) / 256);
        scatter_kernel<<<blocks, 256, 0, stream>>>(ei, wbuf, denom, feat, rst);
    }

    // K7: elu + prediction GEMM (WMMA f32), bias fused
    {
        const int rowTiles = NN / 16;
        pred_gemm_kernel<<<dim3((rowTiles + 7) / 8, MM), 256, 0, stream>>>(rst, Wpred, bpred, out);
    }
}